// RPNModule_23519240913472
// MI455X (gfx1250) — compile-verified
//
#include <hip/hip_runtime.h>
#include <hip/hip_bf16.h>
#include <stdint.h>

// ---------------------------------------------------------------------------
// Problem constants (match reference)
// ---------------------------------------------------------------------------
#define BN       8
#define CC       256
#define HH       80
#define WW       120
#define AA       9
#define HWN      (HH * WW)          // 9600
#define NANCH    (HWN * AA)         // 86400
#define PRE_NMS  750
#define POST_NMS 125
#define NMS_TH   0.7f
#define SEL_N    1024               // padded selection buffer (bitonic friendly)
#define NBINS    1024
#define MROWS    45                 // 9 logits + 36 deltas
#define WPITCH   260                // 256 + 4 pad floats -> bank-conflict-free

typedef __attribute__((ext_vector_type(2))) float  v2f;
typedef __attribute__((ext_vector_type(8))) float  v8f;
typedef __attribute__((ext_vector_type(4))) unsigned int u32x4;
typedef __attribute__((ext_vector_type(8))) int    i32x8;
typedef __attribute__((ext_vector_type(4))) int    i32x4;

// ---------------------------------------------------------------------------
// TDM: build a 2D tensor descriptor (D#) and issue tensor_load_to_lds.
// Group0: count=1 | lds_addr | global_addr[56:0] | type=2.
// Group1: data_size=4B, optional LDS padding (pad_interval code 7 = 256 DW,
//         pad_amount code 3 = 4 DW) -> deposits rows at 260-float pitch.
// This toolchain exposes the 6-arg builtin:
//   (u32x4 g0, i32x8 g1, i32x4 g2, i32x4 g3, i32x8 extra, i32 cpol)
// ---------------------------------------------------------------------------
__device__ __forceinline__ void tdm_load_2d_f32(unsigned lds_byte_addr,
                                                const void* gptr,
                                                unsigned rows,
                                                unsigned row_elems,
                                                unsigned row_stride_elems,
                                                bool pad) {
  unsigned long long ga = (unsigned long long)(uintptr_t)gptr;
  u32x4 g0;
  g0[0] = 1u;                                            // count=1, user D#
  g0[1] = lds_byte_addr;                                 // lds_addr
  g0[2] = (unsigned)(ga & 0xFFFFFFFFu);                  // global_addr lo
  g0[3] = (unsigned)((ga >> 32) & 0x01FFFFFFu) | (2u << 30); // addr hi | type=2

  unsigned d0 = (2u << 16);                              // data_size = 4 bytes
  if (pad) d0 |= (1u << 20) | (7u << 22) | (3u << 25);   // pad 4 DW / 256 DW
  i32x8 g1;
  g1[0] = (int)d0;
  g1[1] = (int)((row_elems & 0xFFFFu) << 16);            // tensor_dim0[15:0]
  g1[2] = (int)(((row_elems >> 16) & 0xFFFFu) |          // tensor_dim0[31:16]
                ((rows & 0xFFFFu) << 16));               // tensor_dim1[15:0]
  g1[3] = (int)(((rows >> 16) & 0xFFFFu) |               // tensor_dim1[31:16]
                ((row_elems & 0xFFFFu) << 16));          // tile_dim0
  g1[4] = (int)(rows & 0xFFFFu);                         // tile_dim1 (tile_dim2=0)
  g1[5] = (int)row_stride_elems;                         // tensor_dim0_stride lo
  g1[6] = 0;                                             // stride hi / dim1_stride
  g1[7] = 0;
  i32x4 z4 = {0, 0, 0, 0};                               // groups 2/3 unused (2D)
  i32x8 z8 = {0, 0, 0, 0, 0, 0, 0, 0};
  __builtin_amdgcn_tensor_load_to_lds(g0, g1, z4, z4, z8, 0);
}

// ---------------------------------------------------------------------------
// Kernel 1: fused GEMM  out[b, o, hw] = sum_c W[o,c] * feats[b,c,hw] + bias[o]
// A = weights (LDS, via TDM), B = feats columns, fp32 WMMA 16x16x4.
// Block = 256 threads (8 waves); wave w owns N-slice of 16 cols + all 3 M tiles.
// Grid = (HW/128, B).
// ---------------------------------------------------------------------------
__global__ __launch_bounds__(256)
void rpn_gemm_wmma(const float* __restrict__ feats,
                   const float* __restrict__ logit_w,
                   const float* __restrict__ logit_b,
                   const float* __restrict__ pred_w,
                   const float* __restrict__ pred_b,
                   float* __restrict__ out_logits,
                   float* __restrict__ out_preds) {
  __shared__ float Wlds[48 * WPITCH];   // 49,920 B, rows 45..47 unused
  __shared__ float Blds[48];            // combined bias

  const int tid   = threadIdx.x;
  const int lane  = tid & 31;
  const int wid   = tid >> 5;
  const int khalf = lane >> 4;          // 0 | 1 (upper half-wave)
  const int b     = blockIdx.y;

  // --- stage weights into LDS with the Tensor Data Mover (wave 0 issues) ---
  if (wid == 0) {
    unsigned base = (unsigned)(uintptr_t)&Wlds[0];
    if (lane == 0) {
      tdm_load_2d_f32(base,                        logit_w,  9u, 256u, 256u, true);
      tdm_load_2d_f32(base + 9u * WPITCH * 4u,     pred_w,  36u, 256u, 256u, true);
    }
    __builtin_amdgcn_s_wait_tensorcnt(0);
  }
  if (tid < 48) Blds[tid] = (tid < 9) ? logit_b[tid]
                           : (tid < 45) ? pred_b[tid - 9] : 0.0f;
  __syncthreads();

  const int n = (blockIdx.x * 8 + wid) * 16 + (lane & 15); // output column
  const float* fcol = feats + (size_t)b * CC * HWN + n;

  v8f acc[3] = {v8f{}, v8f{}, v8f{}};
  const int m = lane & 15;

  for (int k0 = 0; k0 < CC; k0 += 4) {
    // B fragment (4x16): VGPR0 -> rows k0,k0+1 ; VGPR1 -> rows k0+2,k0+3
    v2f bf;
    bf[0] = fcol[(size_t)(k0 + khalf) * HWN];
    bf[1] = fcol[(size_t)(k0 + 2 + khalf) * HWN];
    // A fragments (16x4): lanes 0-15 hold K={k0,k0+1}, lanes 16-31 K={k0+2,k0+3}
    const int ka = k0 + khalf * 2;
    v2f a0, a1, a2;
    a0[0] = Wlds[(m     ) * WPITCH + ka]; a0[1] = Wlds[(m     ) * WPITCH + ka + 1];
    a1[0] = Wlds[(m + 16) * WPITCH + ka]; a1[1] = Wlds[(m + 16) * WPITCH + ka + 1];
    a2[0] = Wlds[(m + 32) * WPITCH + ka]; a2[1] = Wlds[(m + 32) * WPITCH + ka + 1];

    acc[0] = __builtin_amdgcn_wmma_f32_16x16x4_f32(false, a0, false, bf,
                                                   (short)0, acc[0], false, false);
    acc[1] = __builtin_amdgcn_wmma_f32_16x16x4_f32(false, a1, false, bf,
                                                   (short)0, acc[1], false, false);
    acc[2] = __builtin_amdgcn_wmma_f32_16x16x4_f32(false, a2, false, bf,
                                                   (short)0, acc[2], false, false);
  }

  // C/D layout: VGPR r -> row (r + 8*khalf), col = lane&15
  #pragma unroll
  for (int t = 0; t < 3; ++t) {
    #pragma unroll
    for (int r = 0; r < 8; ++r) {
      const int row = t * 16 + r + 8 * khalf;
      if (row < MROWS) {
        const float v = acc[t][r] + Blds[row];
        if (row < 9)
          out_logits[((size_t)b * 9 + row) * HWN + n] = v;
        else
          out_preds[((size_t)b * 36 + (row - 9)) * HWN + n] = v;
      }
    }
  }
}

// ---------------------------------------------------------------------------
// Kernel 0: init workspace (histograms, counters, selection pad)
// ---------------------------------------------------------------------------
__global__ void rpn_init(unsigned* __restrict__ hist,
                         unsigned* __restrict__ ctrA,
                         unsigned* __restrict__ ctrE,
                         float* __restrict__ sel_scores) {
  const int i = blockIdx.x * blockDim.x + threadIdx.x;
  if (i < BN * NBINS) { hist[i] = 0u; sel_scores[i] = -1.0e9f; }
  if (i < BN) { ctrA[i] = 0u; ctrE[i] = 0u; }
}

// ---------------------------------------------------------------------------
// Kernel 2: sigmoid + bbox decode + clip + min-size filter + histogram
// ---------------------------------------------------------------------------
__global__ __launch_bounds__(256)
void rpn_prep(const float* __restrict__ logits,
              const float* __restrict__ preds,
              const float* __restrict__ anchors,
              const float* __restrict__ im_info,
              float* __restrict__ ws_scores,
              float* __restrict__ ws_boxes,
              unsigned* __restrict__ hist) {
  const int idx = blockIdx.x * blockDim.x + threadIdx.x;
  if (idx >= BN * NANCH) return;
  const int b  = idx / NANCH;
  const int i  = idx - b * NANCH;
  const int a  = i % AA;
  const int hw = i / AA;

  float s = logits[((size_t)b * AA + a) * HWN + hw];
  s = 1.0f / (1.0f + __expf(-s));

  const float dx = preds[((size_t)b * 36 + a * 4 + 0) * HWN + hw];
  const float dy = preds[((size_t)b * 36 + a * 4 + 1) * HWN + hw];
  const float dw = preds[((size_t)b * 36 + a * 4 + 2) * HWN + hw];
  const float dh = preds[((size_t)b * 36 + a * 4 + 3) * HWN + hw];

  const float* an = anchors + (size_t)i * 4;
  const float w  = an[2] - an[0] + 1.0f;
  const float h  = an[3] - an[1] + 1.0f;
  const float cx = an[0] + 0.5f * w;
  const float cy = an[1] + 0.5f * h;
  const float px = dx * w + cx;
  const float py = dy * h + cy;
  const float pw = __expf(dw) * w;
  const float ph = __expf(dh) * h;

  const float* info = im_info + b * 4;
  const float xmax = info[1] - 1.0f, ymax = info[0] - 1.0f;
  const float x1 = fminf(fmaxf(px - 0.5f * pw, 0.0f), xmax);
  const float y1 = fminf(fmaxf(py - 0.5f * ph, 0.0f), ymax);
  const float x2 = fminf(fmaxf(px + 0.5f * pw, 0.0f), xmax);
  const float y2 = fminf(fmaxf(py + 0.5f * ph, 0.0f), ymax);

  const bool keep = ((x2 - x1 + 1.0f) >= 16.0f * info[3]) &&
                    ((y2 - y1 + 1.0f) >= 16.0f * info[2]);
  s = keep ? s : -1.0e9f;

  ws_scores[idx] = s;
  float* bx = ws_boxes + (size_t)idx * 4;
  bx[0] = x1; bx[1] = y1; bx[2] = x2; bx[3] = y2;

  unsigned k = __float_as_uint(s);
  k = (k & 0x80000000u) ? ~k : (k | 0x80000000u);   // order-preserving key
  atomicAdd(&hist[b * NBINS + (k >> 22)], 1u);
}

// ---------------------------------------------------------------------------
// Kernel 3: find cutoff bin for top-PRE_NMS per batch
// ---------------------------------------------------------------------------
__global__ void rpn_cutoff(const unsigned* __restrict__ hist,
                           unsigned* __restrict__ cutoff,
                           unsigned* __restrict__ nabove) {
  const int b = blockIdx.x;
  if (threadIdx.x != 0) return;
  unsigned cum = 0;
  unsigned bin = 0, above = 0;
  for (int q = NBINS - 1; q >= 0; --q) {
    const unsigned c = hist[b * NBINS + q];
    if (cum + c >= PRE_NMS || q == 0) { bin = (unsigned)q; above = cum; break; }
    cum += c;
  }
  cutoff[b] = bin;
  nabove[b] = above;
}

// ---------------------------------------------------------------------------
// Kernel 4: compact candidates into 750-slot selection buffer (atomics)
// ---------------------------------------------------------------------------
__global__ __launch_bounds__(256)
void rpn_compact(const float* __restrict__ ws_scores,
                 const float* __restrict__ ws_boxes,
                 const unsigned* __restrict__ cutoff,
                 const unsigned* __restrict__ nabove,
                 unsigned* __restrict__ ctrA,
                 unsigned* __restrict__ ctrE,
                 float* __restrict__ sel_scores,
                 float* __restrict__ sel_boxes) {
  const int idx = blockIdx.x * blockDim.x + threadIdx.x;
  if (idx >= BN * NANCH) return;
  const int b = idx / NANCH;

  const float s = ws_scores[idx];
  unsigned k = __float_as_uint(s);
  k = (k & 0x80000000u) ? ~k : (k | 0x80000000u);
  const unsigned bin = k >> 22;
  const unsigned cb  = cutoff[b];

  unsigned pos;
  if (bin > cb)       pos = atomicAdd(&ctrA[b], 1u);
  else if (bin == cb) pos = nabove[b] + atomicAdd(&ctrE[b], 1u);
  else return;

  if (pos < PRE_NMS) {
    sel_scores[b * SEL_N + pos] = s;
    const float* src = ws_boxes + (size_t)idx * 4;
    float* dst = sel_boxes + ((size_t)b * SEL_N + pos) * 4;
    dst[0] = src[0]; dst[1] = src[1]; dst[2] = src[2]; dst[3] = src[3];
  }
}

// ---------------------------------------------------------------------------
// Kernel 5: per-batch bitonic sort (descending by score), carrying boxes
// ---------------------------------------------------------------------------
__global__ __launch_bounds__(SEL_N)
void rpn_sort(float* __restrict__ sel_scores, float* __restrict__ sel_boxes) {
  __shared__ float ss[SEL_N];
  __shared__ float bx[SEL_N], by[SEL_N], bz[SEL_N], bw[SEL_N];
  const int t = threadIdx.x;
  const int b = blockIdx.x;

  ss[t] = sel_scores[b * SEL_N + t];
  const float* src = sel_boxes + ((size_t)b * SEL_N + t) * 4;
  bx[t] = src[0]; by[t] = src[1]; bz[t] = src[2]; bw[t] = src[3];
  __syncthreads();

  for (int k = 2; k <= SEL_N; k <<= 1) {
    for (int j = k >> 1; j > 0; j >>= 1) {
      const int ixj = t ^ j;
      if (ixj > t) {
        const bool descSeg = ((t & k) == 0);
        const bool doSwap = descSeg ? (ss[t] < ss[ixj]) : (ss[t] > ss[ixj]);
        if (doSwap) {
          float tmp;
          tmp = ss[t]; ss[t] = ss[ixj]; ss[ixj] = tmp;
          tmp = bx[t]; bx[t] = bx[ixj]; bx[ixj] = tmp;
          tmp = by[t]; by[t] = by[ixj]; by[ixj] = tmp;
          tmp = bz[t]; bz[t] = bz[ixj]; bz[ixj] = tmp;
          tmp = bw[t]; bw[t] = bw[ixj]; bw[ixj] = tmp;
        }
      }
      __syncthreads();
    }
  }

  sel_scores[b * SEL_N + t] = ss[t];
  float* dst = sel_boxes + ((size_t)b * SEL_N + t) * 4;
  dst[0] = bx[t]; dst[1] = by[t]; dst[2] = bz[t]; dst[3] = bw[t];
}

// ---------------------------------------------------------------------------
// Kernel 6: NMS (sequential-i, parallel-j) + rank + scatter proposals
// ---------------------------------------------------------------------------
__global__ __launch_bounds__(768)
void rpn_nms(const float* __restrict__ sel_scores,
             const float* __restrict__ sel_boxes,
             float* __restrict__ out_props) {
  __shared__ float X1[PRE_NMS], Y1[PRE_NMS], X2[PRE_NMS], Y2[PRE_NMS], AR[PRE_NMS];
  __shared__ int   keep[PRE_NMS];
  __shared__ int   rank[PRE_NMS];
  const int t = threadIdx.x;
  const int b = blockIdx.x;

  if (t < PRE_NMS) {
    const float* src = sel_boxes + ((size_t)b * SEL_N + t) * 4;
    const float x1 = src[0], y1 = src[1], x2 = src[2], y2 = src[3];
    X1[t] = x1; Y1[t] = y1; X2[t] = x2; Y2[t] = y2;
    AR[t] = (x2 - x1 + 1.0f) * (y2 - y1 + 1.0f);
    keep[t] = sel_scores[b * SEL_N + t] > -1.0e8f ? 1 : 0;
  }
  __syncthreads();

  for (int i = 0; i < PRE_NMS; ++i) {
    if (keep[i] && t > i && t < PRE_NMS && keep[t]) {
      const float xx1 = fmaxf(X1[i], X1[t]);
      const float yy1 = fmaxf(Y1[i], Y1[t]);
      const float xx2 = fminf(X2[i], X2[t]);
      const float yy2 = fminf(Y2[i], Y2[t]);
      const float inter = fmaxf(xx2 - xx1 + 1.0f, 0.0f) *
                          fmaxf(yy2 - yy1 + 1.0f, 0.0f);
      const float iou = inter / (AR[i] + AR[t] - inter);
      if (iou > NMS_TH) keep[t] = 0;
    }
    __syncthreads();
  }

  if (t == 0) {
    int r = 0;
    for (int j = 0; j < PRE_NMS; ++j) { rank[j] = keep[j] ? r : -1; r += keep[j]; }
  }
  float* P = out_props + (size_t)b * POST_NMS * 5;
  for (int q = t; q < POST_NMS * 5; q += blockDim.x) P[q] = 0.0f;
  __syncthreads();

  if (t < PRE_NMS && rank[t] >= 0 && rank[t] < POST_NMS) {
    float* d = P + rank[t] * 5;
    d[0] = (float)b; d[1] = X1[t]; d[2] = Y1[t]; d[3] = X2[t]; d[4] = Y2[t];
  }
}

// ---------------------------------------------------------------------------
// Host launcher
// ---------------------------------------------------------------------------
extern "C" void kernel_launch(void* const* d_in, const int* in_sizes, int n_in,
                              void* d_out, int out_size, void* d_ws, size_t ws_size,
                              hipStream_t stream) {
  const float* feats   = (const float*)d_in[0];
  const float* logit_w = (const float*)d_in[1];
  const float* logit_b = (const float*)d_in[2];
  const float* pred_w  = (const float*)d_in[3];
  const float* pred_b  = (const float*)d_in[4];
  const float* anchors = (const float*)d_in[5];
  const float* im_info = (const float*)d_in[6];

  float* out        = (float*)d_out;
  float* out_logits = out;                                  // 8*9*9600
  float* out_preds  = out + (size_t)BN * 9 * HWN;           // 8*36*9600
  float* out_props  = out + (size_t)BN * MROWS * HWN;       // 8*125*5

  // workspace carve-up
  float* ws_scores  = (float*)d_ws;                         // B*NANCH
  float* ws_boxes   = ws_scores + (size_t)BN * NANCH;       // B*NANCH*4
  float* sel_scores = ws_boxes + (size_t)BN * NANCH * 4;    // B*SEL_N
  float* sel_boxes  = sel_scores + (size_t)BN * SEL_N;      // B*SEL_N*4
  unsigned* hist    = (unsigned*)(sel_boxes + (size_t)BN * SEL_N * 4); // B*NBINS
  unsigned* ctrA    = hist + (size_t)BN * NBINS;            // B
  unsigned* ctrE    = ctrA + BN;                            // B
  unsigned* cutoff  = ctrE + BN;                            // B
  unsigned* nabove  = cutoff + BN;                          // B

  rpn_init<<<(BN * NBINS + 255) / 256, 256, 0, stream>>>(hist, ctrA, ctrE, sel_scores);

  dim3 ggrid(HWN / 128, BN);                                // 75 x 8 blocks
  rpn_gemm_wmma<<<ggrid, 256, 0, stream>>>(feats, logit_w, logit_b,
                                           pred_w, pred_b,
                                           out_logits, out_preds);

  const int nElem = BN * NANCH;
  rpn_prep<<<(nElem + 255) / 256, 256, 0, stream>>>(out_logits, out_preds,
                                                    anchors, im_info,
                                                    ws_scores, ws_boxes, hist);

  rpn_cutoff<<<BN, 64, 0, stream>>>(hist, cutoff, nabove);

  rpn_compact<<<(nElem + 255) / 256, 256, 0, stream>>>(ws_scores, ws_boxes,
                                                       cutoff, nabove,
                                                       ctrA, ctrE,
                                                       sel_scores, sel_boxes);

  rpn_sort<<<BN, SEL_N, 0, stream>>>(sel_scores, sel_boxes);

  rpn_nms<<<BN, 768, 0, stream>>>(sel_scores, sel_boxes, out_props);
}